// HebbianMemory_77386720740104
// MI455X (gfx1250) — compile-verified
//
#include <hip/hip_runtime.h>

// ---------------------------------------------------------------------------
// HebbianMemory on MI455X (gfx1250): chunked linear-attention reformulation.
//   k,v,q = x@W+b           (WMMA f32 GEMMs)
//   scan  -> per-chunk GEMMs: Minter=Q@S, P=Q@K^T, Mintra=mask(P)@V,
//            S' = decay^T*S + (lr*decay-weighted K)^T @ V
//   out   = mems@Wo+bo      (WMMA f32 GEMM)
// State slice (512x64 f32) lives in LDS for the whole sequence (CDNA5 320KB WGP LDS).
// All matmuls use V_WMMA_F32_16X16X4_F32 (fp32 data => fp32 tensor path).
// Recurrence tile loads use GLOBAL_LOAD_ASYNC_TO_LDS_B128 (ASYNCcnt) when the
// toolchain exposes the builtins; otherwise fall back to ld->ds_store.
// ---------------------------------------------------------------------------

#define BATCH   8
#define SEQ     2048
#define DIM     512
#define LR      0.1f
#define DECAY   0.9f
#define CHUNK   64
#define ESLICE  64
#define NCHUNK  (SEQ / CHUNK)
#define SS      (ESLICE + 4)   // LDS row stride (floats): 68 -> bank-conflict free

#if defined(__has_builtin)
#if __has_builtin(__builtin_amdgcn_global_load_async_to_lds_b128) && \
    __has_builtin(__builtin_amdgcn_s_wait_asynccnt)
#define USE_ASYNC_LDS 1
#endif
#endif

typedef float v2f __attribute__((ext_vector_type(2)));
typedef float v8f __attribute__((ext_vector_type(8)));
typedef int   v4i __attribute__((__vector_size__(16)));   // gcc-style, matches builtin

#ifdef USE_ASYNC_LDS
typedef __attribute__((address_space(1))) v4i* gv4i_ptr;  // global int4*
typedef __attribute__((address_space(3))) v4i* lv4i_ptr;  // LDS int4*
#endif

static __device__ __forceinline__ v8f wmma4(v2f a, v2f b, v8f c) {
  // D = A(16x4 f32) x B(4x16 f32) + C(16x16 f32) -> v_wmma_f32_16x16x4_f32
  return __builtin_amdgcn_wmma_f32_16x16x4_f32(false, a, false, b, (short)0, c,
                                               false, false);
}

// Load a 64x64 fp32 tile (row stride ld) from global into LDS (row stride SS).
// 256 threads x 16 floats each. ra = tid>>2, ca = (tid&3)*16.
static __device__ __forceinline__ void ldtile64(const float* __restrict__ g, int ld,
                                                float* s, int ra, int ca) {
#ifdef USE_ASYNC_LDS
  // Direct memory->LDS DMA, tracked with ASYNCcnt; no VGPR staging.
  gv4i_ptr gp = (gv4i_ptr)(g + (size_t)ra * ld + ca);
  lv4i_ptr sp = (lv4i_ptr)(s + ra * SS + ca);
  __builtin_amdgcn_global_load_async_to_lds_b128(gp + 0, sp + 0, 0, 0);
  __builtin_amdgcn_global_load_async_to_lds_b128(gp + 1, sp + 1, 0, 0);
  __builtin_amdgcn_global_load_async_to_lds_b128(gp + 2, sp + 2, 0, 0);
  __builtin_amdgcn_global_load_async_to_lds_b128(gp + 3, sp + 3, 0, 0);
#else
  const float4* gp = (const float4*)(g + (size_t)ra * ld + ca);
  float4 x0 = gp[0], x1 = gp[1], x2 = gp[2], x3 = gp[3];
  float* sp = s + ra * SS + ca;
  ((float4*)sp)[0] = x0; ((float4*)sp)[1] = x1;
  ((float4*)sp)[2] = x2; ((float4*)sp)[3] = x3;
#endif
}

// Wait for this wave's outstanding async tile loads (before barrier-signal).
static __device__ __forceinline__ void tile_wait() {
#ifdef USE_ASYNC_LDS
  __builtin_amdgcn_s_wait_asynccnt(0);
#endif
}

// ---------------------------------------------------------------------------
// Generic C[M x N] = A[M x K] @ W[K x N] + bias[N].  64x64 tile per workgroup,
// 8 waves x two 16x16 accum tiles, K streamed in 32-deep LDS blocks.
// Register-staged double buffering: global loads for block i+1 issue before the
// barrier and overlap block i's WMMAs. Requires M%64==0, N%64==0, K%32==0.
// ---------------------------------------------------------------------------
__global__ __launch_bounds__(256)
void wmma_gemm_bias(const float* __restrict__ A, const float* __restrict__ W,
                    const float* __restrict__ bias, float* __restrict__ C,
                    int K, int N) {
  __shared__ float sA[64 * 36];   // [m][k], stride 36 (pad)
  __shared__ float sB[32 * 68];   // [k][n], stride 68 (pad)

  const int tid  = threadIdx.x;
  const int wave = tid >> 5, lane = tid & 31;
  const int lh = lane & 15, hh = lane >> 4;
  const int mt  = (wave & 3) * 16;
  const int nt0 = ((wave >> 2) * 2) * 16;
  const int nt1 = nt0 + 16;
  const int m0  = blockIdx.x * 64;
  const int n0  = blockIdx.y * 64;

  v8f acc0 = {0.f,0.f,0.f,0.f,0.f,0.f,0.f,0.f};
  v8f acc1 = acc0;

  const int ra = tid >> 2, ca = (tid & 3) * 8;  // A block: 64 rows x 32 cols
  const int rb = tid >> 3, cb = (tid & 7) * 8;  // W block: 32 rows x 64 cols

  for (int kb = 0; kb < K; kb += 32) {
    const float4* ga = (const float4*)(A + (size_t)(m0 + ra) * K + kb + ca);
    float4 a0 = ga[0], a1 = ga[1];
    const float4* gw = (const float4*)(W + (size_t)(kb + rb) * N + n0 + cb);
    float4 w0 = gw[0], w1 = gw[1];
    if (kb + 32 < K) {  // uniform branch; CDNA5 global_prefetch_b8
      __builtin_prefetch(A + (size_t)(m0 + ra) * K + kb + 32 + ca, 0, 0);
      __builtin_prefetch(W + (size_t)(kb + 32 + rb) * N + n0 + cb, 0, 0);
    }
    __syncthreads();
    ((float4*)&sA[ra * 36 + ca])[0] = a0;
    ((float4*)&sA[ra * 36 + ca + 4])[0] = a1;
    ((float4*)&sB[rb * 68 + cb])[0] = w0;
    ((float4*)&sB[rb * 68 + cb + 4])[0] = w1;
    __syncthreads();
#pragma unroll
    for (int k0 = 0; k0 < 32; k0 += 4) {
      const int kk = k0 + hh * 2;
      v2f a;  a.x  = sA[(mt + lh) * 36 + kk];   a.y  = sA[(mt + lh) * 36 + kk + 1];
      v2f b0; b0.x = sB[kk * 68 + nt0 + lh];    b0.y = sB[(kk + 1) * 68 + nt0 + lh];
      v2f b1; b1.x = sB[kk * 68 + nt1 + lh];    b1.y = sB[(kk + 1) * 68 + nt1 + lh];
      acc0 = wmma4(a, b0, acc0);
      acc1 = wmma4(a, b1, acc1);
    }
  }

  const float bi0 = bias[n0 + nt0 + lh];
  const float bi1 = bias[n0 + nt1 + lh];
#pragma unroll
  for (int r = 0; r < 8; ++r) {
    const int row = m0 + mt + r + hh * 8;
    C[(size_t)row * N + n0 + nt0 + lh] = acc0[r] + bi0;
    C[(size_t)row * N + n0 + nt1 + lh] = acc1[r] + bi1;
  }
}

// ---------------------------------------------------------------------------
// Recurrence: one workgroup per (batch, 64-wide e-slice). State slice 512x64 f32
// kept in LDS across all 32 chunks. Per chunk:
//   phase 1: Minter(64x64) += Q@S, P(64x64) += Q@K^T (D streamed in 64-blocks)
//   phase 2: A[t][s] = (s<t) ? lr*decay^(t-1-s)*P[t][s] : 0 ; load V slice
//   phase 3: mem = decay^t * Minter + A@V ; store to mems
//   phase 4: S[d][e] = decay^64*S[d][e] + sum_s (lr*decay^(63-s)K[s][d])*V[s][e]
// ---------------------------------------------------------------------------
__global__ __launch_bounds__(256)
void hebbian_recurrence(const float* __restrict__ Kb, const float* __restrict__ Vb,
                        const float* __restrict__ Qb, float* __restrict__ Mb,
                        float* __restrict__ fstate) {
  extern __shared__ float smem[];
  float* sState = smem;                      // DIM  x SS
  float* sQ     = sState + DIM * SS;         // 64 x SS
  float* sK     = sQ + CHUNK * SS;           // 64 x SS
  float* sV     = sK + CHUNK * SS;           // 64 x SS
  float* sA     = sV + CHUNK * SS;           // 64 x SS
  float* dpow   = sA + CHUNK * SS;           // CHUNK+1 decay powers

  const int tid  = threadIdx.x;
  const int wave = tid >> 5, lane = tid & 31;
  const int lh = lane & 15, hh = lane >> 4;
  const int mt  = (wave & 3) * 16;
  const int nt0 = ((wave >> 2) * 2) * 16;
  const int nt1 = nt0 + 16;

  const int b  = blockIdx.x >> 3;            // DIM/ESLICE == 8 slices
  const int e0 = (blockIdx.x & 7) * ESLICE;

  const float* Kp = Kb + (size_t)b * SEQ * DIM;
  const float* Vp = Vb + (size_t)b * SEQ * DIM;
  const float* Qp = Qb + (size_t)b * SEQ * DIM;
  float*       Mp = Mb + (size_t)b * SEQ * DIM;

  if (tid <= CHUNK) dpow[tid] = powf(DECAY, (float)tid);
  for (int i = tid; i < DIM * SS; i += 256) sState[i] = 0.f;
  __syncthreads();

  const int ra = tid >> 2, ca = (tid & 3) * 16;  // 64x64 tile loader indices
  const v8f vzero = {0.f,0.f,0.f,0.f,0.f,0.f,0.f,0.f};

  for (int c = 0; c < NCHUNK; ++c) {
    const int t0 = c * CHUNK;
    v8f aI0 = vzero, aI1 = vzero;   // Minter tiles (t x e)
    v8f aP0 = vzero, aP1 = vzero;   // P tiles      (t x s)

    // ---- phase 1: stream D in 64-blocks; accumulate Q@S and Q@K^T ----
    for (int db = 0; db < DIM; db += 64) {
      __syncthreads();
      ldtile64(Qp + (size_t)t0 * DIM + db, DIM, sQ, ra, ca);
      ldtile64(Kp + (size_t)t0 * DIM + db, DIM, sK, ra, ca);
      tile_wait();
      __syncthreads();
#pragma unroll
      for (int k0 = 0; k0 < 64; k0 += 4) {
        const int kk = k0 + hh * 2;
        v2f a;  a.x  = sQ[(mt + lh) * SS + kk];  a.y  = sQ[(mt + lh) * SS + kk + 1];
        // B for Minter: S[d][e], K-dim = d
        v2f s0; s0.x = sState[(db + kk) * SS + nt0 + lh];
                s0.y = sState[(db + kk + 1) * SS + nt0 + lh];
        v2f s1; s1.x = sState[(db + kk) * SS + nt1 + lh];
                s1.y = sState[(db + kk + 1) * SS + nt1 + lh];
        // B for P: K^T[d][s] = K[s][d]
        v2f p0; p0.x = sK[(nt0 + lh) * SS + kk]; p0.y = sK[(nt0 + lh) * SS + kk + 1];
        v2f p1; p1.x = sK[(nt1 + lh) * SS + kk]; p1.y = sK[(nt1 + lh) * SS + kk + 1];
        aI0 = wmma4(a, s0, aI0);
        aI1 = wmma4(a, s1, aI1);
        aP0 = wmma4(a, p0, aP0);
        aP1 = wmma4(a, p1, aP1);
      }
    }

    // ---- phase 2: build masked/decayed A from P; load V slice ----
#pragma unroll
    for (int r = 0; r < 8; ++r) {
      const int t  = mt + r + hh * 8;
      const int s0 = nt0 + lh, s1 = nt1 + lh;
      sA[t * SS + s0] = (s0 < t) ? (LR * dpow[t - 1 - s0] * aP0[r]) : 0.f;
      sA[t * SS + s1] = (s1 < t) ? (LR * dpow[t - 1 - s1] * aP1[r]) : 0.f;
    }
    ldtile64(Vp + (size_t)t0 * DIM + e0, DIM, sV, ra, ca);
    tile_wait();
    __syncthreads();

    // ---- phase 3: mem = decay^t * Minter + A@V ; store ----
#pragma unroll
    for (int r = 0; r < 8; ++r) {
      const float dp = dpow[mt + r + hh * 8];
      aI0[r] *= dp; aI1[r] *= dp;
    }
#pragma unroll
    for (int k0 = 0; k0 < 64; k0 += 4) {
      const int kk = k0 + hh * 2;
      v2f a;  a.x  = sA[(mt + lh) * SS + kk];  a.y  = sA[(mt + lh) * SS + kk + 1];
      v2f b0; b0.x = sV[kk * SS + nt0 + lh];   b0.y = sV[(kk + 1) * SS + nt0 + lh];
      v2f b1; b1.x = sV[kk * SS + nt1 + lh];   b1.y = sV[(kk + 1) * SS + nt1 + lh];
      aI0 = wmma4(a, b0, aI0);
      aI1 = wmma4(a, b1, aI1);
    }
#pragma unroll
    for (int r = 0; r < 8; ++r) {
      const int t = mt + r + hh * 8;
      Mp[(size_t)(t0 + t) * DIM + e0 + nt0 + lh] = aI0[r];
      Mp[(size_t)(t0 + t) * DIM + e0 + nt1 + lh] = aI1[r];
    }

    // ---- phase 4: state update, D streamed in 64-blocks ----
    const float dT = dpow[CHUNK];
    for (int db = 0; db < DIM; db += 64) {
      __syncthreads();
      ldtile64(Kp + (size_t)t0 * DIM + db, DIM, sK, ra, ca);
      tile_wait();
      __syncthreads();
      v8f u0, u1;
#pragma unroll
      for (int r = 0; r < 8; ++r) {
        const int d = db + mt + r + hh * 8;
        u0[r] = dT * sState[d * SS + nt0 + lh];
        u1[r] = dT * sState[d * SS + nt1 + lh];
      }
#pragma unroll
      for (int k0 = 0; k0 < 64; k0 += 4) {
        const int kk = k0 + hh * 2;
        const float w0 = LR * dpow[CHUNK - 1 - kk];
        const float w1 = LR * dpow[CHUNK - 2 - kk];
        // A_upd[d][s] = lr*decay^(63-s) * K[s][d]
        v2f a;  a.x  = w0 * sK[kk * SS + mt + lh];
                a.y  = w1 * sK[(kk + 1) * SS + mt + lh];
        v2f b0; b0.x = sV[kk * SS + nt0 + lh];  b0.y = sV[(kk + 1) * SS + nt0 + lh];
        v2f b1; b1.x = sV[kk * SS + nt1 + lh];  b1.y = sV[(kk + 1) * SS + nt1 + lh];
        u0 = wmma4(a, b0, u0);
        u1 = wmma4(a, b1, u1);
      }
#pragma unroll
      for (int r = 0; r < 8; ++r) {
        const int d = db + mt + r + hh * 8;
        sState[d * SS + nt0 + lh] = u0[r];
        sState[d * SS + nt1 + lh] = u1[r];
      }
    }
    __syncthreads();
  }

  // ---- final state slice -> d_out tail ----
  for (int i = tid; i < DIM * ESLICE; i += 256) {
    const int d = i >> 6;      // ESLICE == 64
    const int e = i & 63;
    fstate[(size_t)b * DIM * DIM + (size_t)d * DIM + e0 + e] = sState[d * SS + e];
  }
}

// ---------------------------------------------------------------------------
extern "C" void kernel_launch(void* const* d_in, const int* in_sizes, int n_in,
                              void* d_out, int out_size, void* d_ws, size_t ws_size,
                              hipStream_t stream) {
  (void)in_sizes; (void)n_in; (void)out_size; (void)ws_size;
  const float* x  = (const float*)d_in[0];
  const float* Wk = (const float*)d_in[1];
  const float* bk = (const float*)d_in[2];
  const float* Wv = (const float*)d_in[3];
  const float* bv = (const float*)d_in[4];
  const float* Wq = (const float*)d_in[5];
  const float* bq = (const float*)d_in[6];
  const float* Wo = (const float*)d_in[7];
  const float* bo = (const float*)d_in[8];

  const size_t NB = (size_t)BATCH * SEQ * DIM;   // 8.39M floats per buffer
  float* kbuf = (float*)d_ws;
  float* vbuf = kbuf + NB;
  float* qbuf = vbuf + NB;
  float* mbuf = qbuf + NB;

  float* out    = (float*)d_out;
  float* fstate = out + NB;                      // (B, D, D) tail of d_out

  const int M = BATCH * SEQ;                     // 16384
  dim3 gemmGrid(M / 64, DIM / 64);

  wmma_gemm_bias<<<gemmGrid, 256, 0, stream>>>(x, Wk, bk, kbuf, DIM, DIM);
  wmma_gemm_bias<<<gemmGrid, 256, 0, stream>>>(x, Wv, bv, vbuf, DIM, DIM);
  wmma_gemm_bias<<<gemmGrid, 256, 0, stream>>>(x, Wq, bq, qbuf, DIM, DIM);

  const size_t smemBytes =
      (size_t)(DIM * SS + 4 * CHUNK * SS + CHUNK + 16) * sizeof(float); // ~209 KB
  hebbian_recurrence<<<BATCH * (DIM / ESLICE), 256, smemBytes, stream>>>(
      kbuf, vbuf, qbuf, mbuf, fstate);

  wmma_gemm_bias<<<gemmGrid, 256, 0, stream>>>(mbuf, Wo, bo, out, DIM, DIM);
}